// OptNet_64381559767941
// MI455X (gfx1250) — compile-verified
//
#include <hip/hip_runtime.h>

// ---------------- problem constants ----------------
#define NFEAT 784
#define NHID  1024
#define NCLS  10
#define NCLSP 16      // padded class dim for GEMM2 / P storage
#define NINEQ 200
#define BATCH 8192
#define NITER 25
#define QP_EPSF 1e-4f
#define BN_EPSF 1e-5f
#define SIGMAF  0.1f
#define MU_MINF 1e-6f
#define BIGF    1e30f

typedef float v2f __attribute__((ext_vector_type(2)));
typedef float v8f __attribute__((ext_vector_type(8)));

#define LIDX(j,k) (((j)*((j)+1))/2 + (k))   // lower-triangular packed index, j>=k

// =====================================================================
// GEMM1: A1 = relu(x @ W1^T + b1)   [8192 x 1024], fp32 WMMA 16x16x4.
// Wave computes 32(M) x 64(N): 8 independent accumulator chains,
// 2 A-frag + 4 B-frag b64 loads per 8 WMMAs (0.75 loads/WMMA).
// A frag (16x4 f32): lanes 0-15 hold K={k,k+1}, lanes 16-31 K={k+2,k+3}
// -> one float2 per lane from a contiguous x row.
// B frag (4x16 f32): lane r = output col, same K split -> float2 from W1 row.
// C/D frag: acc[v] = C[M = v + 8*half][N = r].
// =====================================================================
__global__ __launch_bounds__(256) void k_gemm1(
    const float* __restrict__ x, const float* __restrict__ W1,
    const float* __restrict__ bias1, float* __restrict__ A1)
{
  const int lane = threadIdx.x & 31;
  const int wave = threadIdx.x >> 5;
  const int r    = lane & 15;
  const int half = lane >> 4;
  const int m0 = (blockIdx.y * 8 + wave) * 32;
  const int n0 = blockIdx.x * 64;

  const float* xp0 = x + (size_t)(m0 + r) * NFEAT + 2 * half;
  const float* xp1 = xp0 + (size_t)16 * NFEAT;
  const float* wp  = W1 + (size_t)(n0 + r) * NFEAT + 2 * half;

  v8f acc[8];
  #pragma unroll
  for (int i = 0; i < 8; ++i) acc[i] = {};

  for (int k = 0; k < NFEAT; k += 4) {            // 196 K-steps
    v2f a0  = *(const v2f*)(xp0 + k);
    v2f a1  = *(const v2f*)(xp1 + k);
    v2f bf0 = *(const v2f*)(wp + k);
    v2f bf1 = *(const v2f*)(wp + 16 * NFEAT + k);
    v2f bf2 = *(const v2f*)(wp + 32 * NFEAT + k);
    v2f bf3 = *(const v2f*)(wp + 48 * NFEAT + k);
    acc[0] = __builtin_amdgcn_wmma_f32_16x16x4_f32(false, a0, false, bf0, (short)0, acc[0], false, false);
    acc[1] = __builtin_amdgcn_wmma_f32_16x16x4_f32(false, a0, false, bf1, (short)0, acc[1], false, false);
    acc[2] = __builtin_amdgcn_wmma_f32_16x16x4_f32(false, a0, false, bf2, (short)0, acc[2], false, false);
    acc[3] = __builtin_amdgcn_wmma_f32_16x16x4_f32(false, a0, false, bf3, (short)0, acc[3], false, false);
    acc[4] = __builtin_amdgcn_wmma_f32_16x16x4_f32(false, a1, false, bf0, (short)0, acc[4], false, false);
    acc[5] = __builtin_amdgcn_wmma_f32_16x16x4_f32(false, a1, false, bf1, (short)0, acc[5], false, false);
    acc[6] = __builtin_amdgcn_wmma_f32_16x16x4_f32(false, a1, false, bf2, (short)0, acc[6], false, false);
    acc[7] = __builtin_amdgcn_wmma_f32_16x16x4_f32(false, a1, false, bf3, (short)0, acc[7], false, false);
  }

  #pragma unroll
  for (int mt = 0; mt < 2; ++mt) {
    #pragma unroll
    for (int t = 0; t < 4; ++t) {
      const int col = n0 + t * 16 + r;
      const float bv = bias1[col];
      #pragma unroll
      for (int v = 0; v < 8; ++v) {
        const int row = m0 + mt * 16 + v + 8 * half;
        A1[(size_t)row * NHID + col] = fmaxf(acc[mt * 4 + t][v] + bv, 0.0f);
      }
    }
  }
}

// =====================================================================
// BN1 stats: per-feature biased mean/var over batch -> scale/shift
// =====================================================================
__global__ __launch_bounds__(256) void k_bnstats1(
    const float* __restrict__ A1, const float* __restrict__ gamma,
    const float* __restrict__ beta, float* __restrict__ scale,
    float* __restrict__ shift)
{
  __shared__ float ssum[256], ssq[256];
  const int col = blockIdx.x * 64 + (threadIdx.x & 63);
  const int rp  = threadIdx.x >> 6;   // 4 row partials
  float sum = 0.f, sq = 0.f;
  for (int rw = rp; rw < BATCH; rw += 4) {
    const float v = A1[(size_t)rw * NHID + col];
    sum += v; sq += v * v;
  }
  ssum[threadIdx.x] = sum; ssq[threadIdx.x] = sq;
  __syncthreads();
  if (rp == 0) {
    const int i = threadIdx.x;
    sum = ssum[i] + ssum[i + 64] + ssum[i + 128] + ssum[i + 192];
    sq  = ssq[i]  + ssq[i + 64]  + ssq[i + 128]  + ssq[i + 192];
    const float m   = sum * (1.0f / BATCH);
    const float var = sq * (1.0f / BATCH) - m * m;
    const float sc  = gamma[col] * rsqrtf(var + BN_EPSF);
    scale[col] = sc;
    shift[col] = beta[col] - m * sc;
  }
}

// =====================================================================
// Fold BN1 into W2: W2f[n,k] = W2[n,k]*scale1[k] (rows 10..15 = 0),
// b2f[n] = b2[n] + sum_k W2[n,k]*shift1[k].
// Makes GEMM2's inner loop pure loads + WMMA.
// =====================================================================
__global__ __launch_bounds__(256) void k_fold(
    const float* __restrict__ W2, const float* __restrict__ b2,
    const float* __restrict__ scale1, const float* __restrict__ shift1,
    float* __restrict__ W2f, float* __restrict__ b2f)
{
  __shared__ float red[256];
  const int n = blockIdx.x;        // 0..15
  float partial = 0.f;
  if (n < NCLS) {
    for (int k = threadIdx.x; k < NHID; k += 256) {
      const float w = W2[(size_t)n * NHID + k];
      W2f[(size_t)n * NHID + k] = w * scale1[k];
      partial += w * shift1[k];
    }
  } else {
    for (int k = threadIdx.x; k < NHID; k += 256)
      W2f[(size_t)n * NHID + k] = 0.f;
  }
  red[threadIdx.x] = partial;
  __syncthreads();
  for (int off = 128; off > 0; off >>= 1) {
    if (threadIdx.x < off) red[threadIdx.x] += red[threadIdx.x + off];
    __syncthreads();
  }
  if (threadIdx.x == 0)
    b2f[n] = (n < NCLS ? b2[n] : 0.0f) + red[0];
}

// =====================================================================
// GEMM2: P = relu( A1 @ W2f^T + b2f )  [8192 x 16] (BN1 pre-folded).
// Wave computes 32(M) x 16(N): 3 b64 loads per 2 WMMAs, no VALU in loop.
// =====================================================================
__global__ __launch_bounds__(256) void k_gemm2(
    const float* __restrict__ A1, const float* __restrict__ W2f,
    const float* __restrict__ b2f, float* __restrict__ P)
{
  const int lane = threadIdx.x & 31;
  const int wave = threadIdx.x >> 5;
  const int r    = lane & 15;
  const int half = lane >> 4;
  const int m0   = (blockIdx.x * 8 + wave) * 32;

  const float* ap0 = A1 + (size_t)(m0 + r) * NHID + 2 * half;
  const float* ap1 = ap0 + (size_t)16 * NHID;
  const float* wp  = W2f + (size_t)r * NHID + 2 * half;   // padded rows are zero

  v8f acc0 = {}, acc1 = {};
  for (int k = 0; k < NHID; k += 4) {             // 256 K-steps
    v2f a0 = *(const v2f*)(ap0 + k);
    v2f a1 = *(const v2f*)(ap1 + k);
    v2f b  = *(const v2f*)(wp + k);
    acc0 = __builtin_amdgcn_wmma_f32_16x16x4_f32(false, a0, false, b, (short)0, acc0, false, false);
    acc1 = __builtin_amdgcn_wmma_f32_16x16x4_f32(false, a1, false, b, (short)0, acc1, false, false);
  }
  const float bv = b2f[r];
  #pragma unroll
  for (int v = 0; v < 8; ++v) {
    const int row0 = m0 + v + 8 * half;
    P[(size_t)row0 * NCLSP + r]        = fmaxf(acc0[v] + bv, 0.0f);
    P[(size_t)(row0 + 16) * NCLSP + r] = fmaxf(acc1[v] + bv, 0.0f);
  }
}

// =====================================================================
// BN2 stats over the 10 valid columns of P
// =====================================================================
__global__ __launch_bounds__(256) void k_bnstats2(
    const float* __restrict__ P, const float* __restrict__ gamma,
    const float* __restrict__ beta, float* __restrict__ scale,
    float* __restrict__ shift)
{
  __shared__ float ssum[256], ssq[256];
  const int col = blockIdx.x;   // 0..9
  float sum = 0.f, sq = 0.f;
  for (int rw = threadIdx.x; rw < BATCH; rw += 256) {
    const float v = P[(size_t)rw * NCLSP + col];
    sum += v; sq += v * v;
  }
  ssum[threadIdx.x] = sum; ssq[threadIdx.x] = sq;
  __syncthreads();
  for (int off = 128; off > 0; off >>= 1) {
    if (threadIdx.x < off) {
      ssum[threadIdx.x] += ssum[threadIdx.x + off];
      ssq[threadIdx.x]  += ssq[threadIdx.x + off];
    }
    __syncthreads();
  }
  if (threadIdx.x == 0) {
    const float m   = ssum[0] * (1.0f / BATCH);
    const float var = ssq[0] * (1.0f / BATCH) - m * m;
    const float sc  = gamma[col] * rsqrtf(var + BN_EPSF);
    scale[col] = sc;
    shift[col] = beta[col] - m * sc;
  }
}

// =====================================================================
// Prep: Q = tril(L) tril(L)^T + eps I ; chol(Q) ; h = G z0 + s0
// =====================================================================
__global__ void k_prep(const float* __restrict__ L, const float* __restrict__ G,
                       const float* __restrict__ z0, const float* __restrict__ s0,
                       float* __restrict__ Qout, float* __restrict__ cholOut,
                       float* __restrict__ hOut)
{
  __shared__ float sQ[NCLS * NCLS];
  const int tid = threadIdx.x;
  if (tid < NCLS * NCLS) {
    const int i = tid / NCLS, j = tid % NCLS;
    const int mn = i < j ? i : j;
    float acc = (i == j) ? QP_EPSF : 0.0f;
    for (int u = 0; u <= mn; ++u) acc += L[i * NCLS + u] * L[j * NCLS + u];
    sQ[tid] = acc;
    Qout[tid] = acc;
  }
  if (tid < NINEQ) {
    float acc = s0[tid];
    for (int j = 0; j < NCLS; ++j) acc += G[tid * NCLS + j] * z0[j];
    hOut[tid] = acc;
  }
  __syncthreads();
  if (tid == 0) {
    float A[55];
    for (int i = 0; i < NCLS; ++i)
      for (int j = 0; j <= i; ++j) A[LIDX(i, j)] = sQ[i * NCLS + j];
    for (int k = 0; k < NCLS; ++k) {
      float d = A[LIDX(k, k)];
      for (int u = 0; u < k; ++u) d -= A[LIDX(k, u)] * A[LIDX(k, u)];
      const float dg = sqrtf(fmaxf(d, 1e-20f));
      const float iv = 1.0f / dg;
      A[LIDX(k, k)] = dg;
      for (int j = k + 1; j < NCLS; ++j) {
        float v = A[LIDX(j, k)];
        for (int u = 0; u < k; ++u) v -= A[LIDX(j, u)] * A[LIDX(k, u)];
        A[LIDX(j, k)] = v * iv;
      }
    }
    for (int i = 0; i < 55; ++i) cholOut[i] = A[i];
  }
}

// =====================================================================
// QP interior point, one thread per batch sample. s/lam live in LDS
// (thread-major -> conflict-free). H built as 55-entry lower triangle in
// registers; fully unrolled 10x10 Cholesky (constant indices, no spill).
// Ends with fused log_softmax.
// =====================================================================
#define QPT 32

__global__ __launch_bounds__(QPT) void k_qp(
    const float* __restrict__ P, const float* __restrict__ scale2,
    const float* __restrict__ shift2, const float* __restrict__ Qm,
    const float* __restrict__ cholQ, const float* __restrict__ hvec,
    const float* __restrict__ Gm, float* __restrict__ out)
{
  __shared__ float sG[NINEQ * NCLS];   // 8000 B
  __shared__ float sQ[NCLS * NCLS];    // 400 B
  __shared__ float sC[55];             // chol(Q)
  __shared__ float sHv[NINEQ];         // h
  __shared__ float sS[NINEQ][QPT];     // 25600 B
  __shared__ float sLam[NINEQ][QPT];   // 25600 B

  const int t = threadIdx.x;
  for (int i = t; i < NINEQ * NCLS; i += QPT) sG[i] = Gm[i];
  for (int i = t; i < NCLS * NCLS;  i += QPT) sQ[i] = Qm[i];
  for (int i = t; i < 55;           i += QPT) sC[i] = cholQ[i];
  for (int i = t; i < NINEQ;        i += QPT) sHv[i] = hvec[i];
  __syncthreads();

  const int b = blockIdx.x * QPT + t;

  float p[NCLS], z[NCLS];
  #pragma unroll
  for (int j = 0; j < NCLS; ++j)
    p[j] = P[(size_t)b * NCLSP + j] * scale2[j] + shift2[j];

  { // z = -Q^{-1} p via precomputed chol(Q)
    float ivq[NCLS];
    #pragma unroll
    for (int j = 0; j < NCLS; ++j) ivq[j] = 1.0f / sC[LIDX(j, j)];
    float y[NCLS];
    #pragma unroll
    for (int j = 0; j < NCLS; ++j) {
      float v = -p[j];
      #pragma unroll
      for (int k = 0; k < j; ++k) v -= sC[LIDX(j, k)] * y[k];
      y[j] = v * ivq[j];
    }
    #pragma unroll
    for (int j = NCLS - 1; j >= 0; --j) {
      float v = y[j];
      #pragma unroll
      for (int k = j + 1; k < NCLS; ++k) v -= sC[LIDX(k, j)] * z[k];
      z[j] = v * ivq[j];
    }
  }

  for (int c = 0; c < NINEQ; ++c) {      // s = max(h - Gz, 1), lam = 1
    float gz = 0.f;
    #pragma unroll
    for (int j = 0; j < NCLS; ++j) gz += sG[c * NCLS + j] * z[j];
    const float sv = sHv[c] - gz;
    sS[c][t]   = sv > 1.0f ? sv : 1.0f;
    sLam[c][t] = 1.0f;
  }

  for (int it = 0; it < NITER; ++it) {
    // ---- pass 1: mu ----
    float musum = 0.f;
    for (int c = 0; c < NINEQ; ++c) musum += sS[c][t] * sLam[c][t];
    float mu = musum * (1.0f / NINEQ);
    mu = mu > MU_MINF ? mu : MU_MINF;
    const float smu = SIGMAF * mu;

    // ---- rhs = -(Qz + p) + sum_c wc*G[c]; H = Q + G^T D G ----
    float rhs[NCLS];
    #pragma unroll
    for (int j = 0; j < NCLS; ++j) {
      float v = p[j];
      #pragma unroll
      for (int k = 0; k < NCLS; ++k) v += sQ[j * NCLS + k] * z[k]; // Q symmetric
      rhs[j] = -v;
    }
    float Hm[55];
    #pragma unroll
    for (int i = 0; i < 55; ++i) Hm[i] = 0.f;

    for (int c = 0; c < NINEQ; ++c) {
      float g[NCLS];
      #pragma unroll
      for (int j = 0; j < NCLS; ++j) g[j] = sG[c * NCLS + j];
      float gz = 0.f;
      #pragma unroll
      for (int j = 0; j < NCLS; ++j) gz += g[j] * z[j];
      const float sc   = sS[c][t];
      const float lc   = sLam[c][t];
      const float invs = 1.0f / sc;
      const float D    = lc * invs;
      const float rp   = gz + sc - sHv[c];
      const float wc   = -(smu * invs + D * rp);
      #pragma unroll
      for (int j = 0; j < NCLS; ++j) rhs[j] += wc * g[j];
      #pragma unroll
      for (int j = 0; j < NCLS; ++j) {
        const float gjD = D * g[j];
        #pragma unroll
        for (int k = 0; k <= j; ++k) Hm[LIDX(j, k)] += gjD * g[k];
      }
    }
    #pragma unroll
    for (int j = 0; j < NCLS; ++j) {
      #pragma unroll
      for (int k = 0; k <= j; ++k) Hm[LIDX(j, k)] += sQ[j * NCLS + k];
    }

    // ---- Cholesky of H in place + solve H dz = rhs ----
    float ivd[NCLS];
    #pragma unroll
    for (int k = 0; k < NCLS; ++k) {
      float d = Hm[LIDX(k, k)];
      #pragma unroll
      for (int u = 0; u < k; ++u) d -= Hm[LIDX(k, u)] * Hm[LIDX(k, u)];
      d = d > 1e-20f ? d : 1e-20f;
      const float dg = sqrtf(d);
      const float iv = 1.0f / dg;
      Hm[LIDX(k, k)] = dg; ivd[k] = iv;
      #pragma unroll
      for (int j = k + 1; j < NCLS; ++j) {
        float v = Hm[LIDX(j, k)];
        #pragma unroll
        for (int u = 0; u < k; ++u) v -= Hm[LIDX(j, u)] * Hm[LIDX(k, u)];
        Hm[LIDX(j, k)] = v * iv;
      }
    }
    float dz[NCLS];
    {
      float y[NCLS];
      #pragma unroll
      for (int j = 0; j < NCLS; ++j) {
        float v = rhs[j];
        #pragma unroll
        for (int k = 0; k < j; ++k) v -= Hm[LIDX(j, k)] * y[k];
        y[j] = v * ivd[j];
      }
      #pragma unroll
      for (int j = NCLS - 1; j >= 0; --j) {
        float v = y[j];
        #pragma unroll
        for (int k = j + 1; k < NCLS; ++k) v -= Hm[LIDX(k, j)] * dz[k];
        dz[j] = v * ivd[j];
      }
    }

    // ---- pass 3: fraction-to-boundary step length ----
    float astep = BIGF;
    for (int c = 0; c < NINEQ; ++c) {
      float gz = 0.f, gdz = 0.f;
      #pragma unroll
      for (int j = 0; j < NCLS; ++j) {
        const float gj = sG[c * NCLS + j];
        gz += gj * z[j]; gdz += gj * dz[j];
      }
      const float sc   = sS[c][t];
      const float lc   = sLam[c][t];
      const float invs = 1.0f / sc;
      const float rp   = gz + sc - sHv[c];
      const float ds   = -rp - gdz;
      const float dl   = -lc + smu * invs - (lc * invs) * ds;
      if (ds < 0.f) { const float cand = -sc / ds; astep = astep < cand ? astep : cand; }
      if (dl < 0.f) { const float cand = -lc / dl; astep = astep < cand ? astep : cand; }
    }
    float alpha = 0.99f * astep;
    alpha = alpha < 1.0f ? alpha : 1.0f;

    // ---- pass 4: apply update (recompute ds/dl; cheaper than storing) ----
    for (int c = 0; c < NINEQ; ++c) {
      float gz = 0.f, gdz = 0.f;
      #pragma unroll
      for (int j = 0; j < NCLS; ++j) {
        const float gj = sG[c * NCLS + j];
        gz += gj * z[j]; gdz += gj * dz[j];
      }
      const float sc   = sS[c][t];
      const float lc   = sLam[c][t];
      const float invs = 1.0f / sc;
      const float rp   = gz + sc - sHv[c];
      const float ds   = -rp - gdz;
      const float dl   = -lc + smu * invs - (lc * invs) * ds;
      sS[c][t]   = sc + alpha * ds;
      sLam[c][t] = lc + alpha * dl;
    }
    #pragma unroll
    for (int j = 0; j < NCLS; ++j) z[j] += alpha * dz[j];
  }

  // ---- fused log_softmax ----
  float mx = z[0];
  #pragma unroll
  for (int j = 1; j < NCLS; ++j) mx = mx > z[j] ? mx : z[j];
  float se = 0.f;
  #pragma unroll
  for (int j = 0; j < NCLS; ++j) se += expf(z[j] - mx);
  const float lse = logf(se);
  #pragma unroll
  for (int j = 0; j < NCLS; ++j)
    out[(size_t)b * NCLS + j] = z[j] - mx - lse;
}

// =====================================================================
extern "C" void kernel_launch(void* const* d_in, const int* in_sizes, int n_in,
                              void* d_out, int out_size, void* d_ws, size_t ws_size,
                              hipStream_t stream)
{
  const float* x      = (const float*)d_in[0];
  const float* W1     = (const float*)d_in[1];
  const float* b1     = (const float*)d_in[2];
  const float* gamma1 = (const float*)d_in[3];
  const float* beta1  = (const float*)d_in[4];
  const float* W2     = (const float*)d_in[5];
  const float* b2     = (const float*)d_in[6];
  const float* gamma2 = (const float*)d_in[7];
  const float* beta2  = (const float*)d_in[8];
  const float* L      = (const float*)d_in[9];
  const float* G      = (const float*)d_in[10];
  const float* z0     = (const float*)d_in[11];
  const float* s0     = (const float*)d_in[12];
  float* out = (float*)d_out;

  // workspace layout (floats):
  // A1 | P | scale1 | shift1 | scale2 | shift2 | Q | cholQ | h | W2f | b2f
  float* ws     = (float*)d_ws;
  float* A1     = ws;                                  // 8192*1024
  float* P      = A1 + (size_t)BATCH * NHID;           // 8192*16
  float* scale1 = P + (size_t)BATCH * NCLSP;           // 1024
  float* shift1 = scale1 + NHID;                       // 1024
  float* scale2 = shift1 + NHID;                       // 16
  float* shift2 = scale2 + NCLSP;                      // 16
  float* Qm     = shift2 + NCLSP;                      // 100
  float* cholQ  = Qm + NCLS * NCLS;                    // 55 (pad 64)
  float* hvec   = cholQ + 64;                          // 200 (pad 256)
  float* W2f    = hvec + 256;                          // 16*1024
  float* b2f    = W2f + (size_t)NCLSP * NHID;          // 16

  k_gemm1   <<<dim3(NHID / 64, BATCH / 256), 256, 0, stream>>>(x, W1, b1, A1);
  k_bnstats1<<<NHID / 64, 256, 0, stream>>>(A1, gamma1, beta1, scale1, shift1);
  k_fold    <<<NCLSP, 256, 0, stream>>>(W2, b2, scale1, shift1, W2f, b2f);
  k_gemm2   <<<BATCH / 256, 256, 0, stream>>>(A1, W2f, b2f, P);
  k_bnstats2<<<NCLS, 256, 0, stream>>>(P, gamma2, beta2, scale2, shift2);
  k_prep    <<<1, 256, 0, stream>>>(L, G, z0, s0, Qm, cholQ, hvec);
  k_qp      <<<BATCH / QPT, QPT, 0, stream>>>(P, scale2, shift2, Qm, cholQ, hvec, G, out);

  (void)in_sizes; (void)n_in; (void)out_size; (void)ws_size;
}